// Fetch_Module_28587302323004
// MI455X (gfx1250) — compile-verified
//
#include <hip/hip_runtime.h>
#include <stdint.h>

// Bilinear disparity warp (grid_sample, bilinear, align_corners=False, zeros pad)
//   ix = (x - disp[b,y,x]) * W/(W-1) - 0.5
//   iy = y * H/(H-1) - 0.5
// Memory-bound (~354 MB compulsory @ 23.3 TB/s ~ 15 us).
//
// CDNA5 strategy:
//  - One block per (row y, batch b); 320 threads (10 wave32 waves), 4 px/thread.
//  - Stage the two source rows into zero-guarded padded LDS rows with
//    global_load_async_to_lds_b128 (ASYNCcnt), overlapped with NT disp load.
//  - Guard zeros implement 'zeros' padding -> maskless inner loop; adjacent
//    taps P[k],P[k+1] share a base -> ds_load_2addr_b32 (2 DS ops/pixel).
//  - Row-uniform vertical weights/masks scalarized (s_floor_f32 etc.).
//  - NT b128 store of the output quad.

typedef float v4f __attribute__((ext_vector_type(4)));

#define IMG_W 1280
#define IMG_H 720
#define PADL  4                      // left pad (keeps row start 16B-aligned)
#define PROW  (PADL + IMG_W + 4)     // 1288 floats = 5152 B (16B multiple)

__device__ __forceinline__ uint32_t lds_addr(const void* p) {
    // Low 32 bits of a generic shared-aperture address are the LDS byte offset
    // (see ISA 10.2).
    return (uint32_t)(uintptr_t)p;
}

__global__ __launch_bounds__(320) void warp_bilinear_kernel(
    const float* __restrict__ img,   // [B,1,H,W]
    const float* __restrict__ disp,  // [B,1,H,W]
    float* __restrict__ out)         // [B,1,H,W]
{
    __shared__ __align__(16) float P[2][PROW];   // 2 padded rows, ~10 KB LDS

    const int t = threadIdx.x;                   // 0..319 : quad index in row
    const int y = blockIdx.x;                    // 0..719
    const int b = blockIdx.y;                    // 0..31

    const float SX = (float)IMG_W / (float)(IMG_W - 1);
    const float SY = (float)IMG_H / (float)(IMG_H - 1);

    // ---- vertical sample position: uniform per block (scalarized) ----
    const float iy  = (float)y * SY - 0.5f;
    const float y0f = floorf(iy);
    const float wy1 = iy - y0f;
    const float wy0 = 1.0f - wy1;
    const int   y0  = (int)y0f;
    const int   y1  = y0 + 1;
    const float wy0m = (y0 >= 0 && y0 < IMG_H) ? wy0 : 0.0f;  // mask folded in
    const float wy1m = (y1 >= 0 && y1 < IMG_H) ? wy1 : 0.0f;
    const int   y0c = min(max(y0, 0), IMG_H - 1);
    const int   y1c = min(max(y1, 0), IMG_H - 1);

    const float* __restrict__ im = img + (size_t)b * (IMG_H * (size_t)IMG_W);

    // ---- zero the guard cells (plain ds_store; ordered by the barrier) ----
    if (t < 4) {
        P[0][t] = 0.0f;  P[0][PADL + IMG_W + t] = 0.0f;   // P[2],P[3] & P[1284] used
        P[1][t] = 0.0f;  P[1][PADL + IMG_W + t] = 0.0f;
    }

    // ---- async-stage rows y0c,y1c into P[r][4..1283]: one b128/thread/row ----
    const uint32_t l0 = lds_addr(&P[0][PADL]) + (uint32_t)t * 16u;
    const uint32_t l1 = lds_addr(&P[1][PADL]) + (uint32_t)t * 16u;
    const char* g0 = (const char*)(im + (size_t)y0c * IMG_W) + (size_t)t * 16;
    const char* g1 = (const char*)(im + (size_t)y1c * IMG_W) + (size_t)t * 16;
    asm volatile("global_load_async_to_lds_b128 %0, %1, off"
                 :: "v"(l0), "v"(g0) : "memory");
    asm volatile("global_load_async_to_lds_b128 %0, %1, off"
                 :: "v"(l1), "v"(g1) : "memory");

    // ---- overlap: NT b128 load of this thread's 4 disparities ----
    const size_t rowoff = ((size_t)b * IMG_H + (size_t)y) * IMG_W;
    const v4f d4 = __builtin_nontemporal_load((const v4f*)(disp + rowoff) + t);

    // ---- publish LDS: wait own async copies, then workgroup barrier ----
    asm volatile("s_wait_asynccnt 0x0" ::: "memory");
    __syncthreads();

    // ---- 4 pixels: maskless horizontal lerp via guard zeros, vertical blend ----
    const float dv[4] = {d4.x, d4.y, d4.z, d4.w};
    float res[4];
#pragma unroll
    for (int i = 0; i < 4; ++i) {
        const int   x   = t * 4 + i;
        const float ix  = ((float)x - dv[i]) * SX - 0.5f;
        const float x0f = floorf(ix);
        const float wx1 = ix - x0f;
        const float wx0 = 1.0f - wx1;
        const int   x0  = (int)x0f;          // in [-66, 1279]
        const int   k   = PADL + max(x0, -2); // >= 2; OOB taps land on guard zeros

        const float a0 = P[0][k], a1 = P[0][k + 1];   // ds_load_2addr_b32
        const float b0 = P[1][k], b1 = P[1][k + 1];   // ds_load_2addr_b32
        res[i] = (a0 * wx0 + a1 * wx1) * wy0m
               + (b0 * wx0 + b1 * wx1) * wy1m;
    }

    v4f o;
    o.x = res[0]; o.y = res[1]; o.z = res[2]; o.w = res[3];
    __builtin_nontemporal_store(o, (v4f*)(out + rowoff) + t);
}

extern "C" void kernel_launch(void* const* d_in, const int* in_sizes, int n_in,
                              void* d_out, int out_size, void* d_ws, size_t ws_size,
                              hipStream_t stream) {
    (void)in_sizes; (void)n_in; (void)d_ws; (void)ws_size; (void)out_size;
    const float* img  = (const float*)d_in[0];   // right_img [B,1,H,W] f32
    const float* disp = (const float*)d_in[1];   // disp      [B,1,H,W] f32
    float* out = (float*)d_out;                  // [B,1,H,W] f32

    dim3 grid(IMG_H, 32, 1);    // (y, b)
    dim3 block(320, 1, 1);      // W/4 quads, 10 wave32 waves
    warp_bilinear_kernel<<<grid, block, 0, stream>>>(img, disp, out);
}